// GINE_53197464928922
// MI455X (gfx1250) — compile-verified
//
#include <hip/hip_runtime.h>

#define D 64

typedef __attribute__((ext_vector_type(2))) float v2f;
typedef __attribute__((ext_vector_type(8))) float v8f;

// ---------------------------------------------------------------------------
// h[n,d] = feat_emb[x[n], d]
// ---------------------------------------------------------------------------
__global__ __launch_bounds__(256) void k_embed(const int* __restrict__ x,
                                               const float* __restrict__ emb,
                                               float* __restrict__ h, int total) {
  int i = blockIdx.x * 256 + threadIdx.x;
  if (i < total) {
    int n = i >> 6, d = i & 63;
    h[i] = emb[x[n] * D + d];
  }
}

// ---------------------------------------------------------------------------
// msg = relu(h[src] + edge_emb[eattr]); agg[dst] += msg   (atomic scatter)
// 64 threads per edge, edge_emb table staged in LDS.
// ---------------------------------------------------------------------------
__global__ __launch_bounds__(256) void k_message(
    const float* __restrict__ h, const int* __restrict__ src,
    const int* __restrict__ dst, const int* __restrict__ ea,
    const float* __restrict__ eemb, float* __restrict__ agg, int totalED, int NEv) {
  __shared__ float se[4 * D];
  int tid = threadIdx.x;
  if (tid < NEv * D) se[tid] = eemb[tid];
  __syncthreads();
  int i = blockIdx.x * 256 + tid;
  if (i < totalED) {
    int e = i >> 6, d = i & 63;
    float m = h[src[e] * D + d] + se[ea[e] * D + d];
    if (m > 0.0f) unsafeAtomicAdd(&agg[dst[e] * D + d], m);
  }
}

// ---------------------------------------------------------------------------
// pass1: z = (1+eps)*h + agg ;  y = z @ W1  (bias absorbed by BN)
// One wave per 16-node tile, WMMA f32 16x16x4, W1 staged in LDS.
// Also accumulates per-column sum / sumsq of y for BN statistics.
// ---------------------------------------------------------------------------
__global__ __launch_bounds__(256) void k_pass1(
    const float* __restrict__ h, const float* __restrict__ agg,
    const float* __restrict__ W, const float* __restrict__ epsArr, int layer,
    float* __restrict__ y, float* __restrict__ colSum, float* __restrict__ colSq,
    int nNodes) {
  __shared__ float sW[D * D];
  __shared__ float sSum[D];
  __shared__ float sSq[D];
  int tid = threadIdx.x;
  for (int i = tid; i < D * D; i += 256) sW[i] = W[i];
  if (tid < D) { sSum[tid] = 0.0f; sSq[tid] = 0.0f; }
  __syncthreads();

  float epsv = 1.0f + epsArr[layer];
  int lane  = tid & 31;
  int wave  = tid >> 5;
  int mrow  = lane & 15;            // A-matrix row (both half-waves cover M=0..15)
  int khalf = (lane >> 4) << 1;     // K offset: 0 for lanes 0-15, 2 for lanes 16-31
  int nbase = blockIdx.x * 128 + wave * 16;

  if (nbase < nNodes) {             // wave-uniform: EXEC stays all-ones for WMMA
    const float* hrow = h   + (nbase + mrow) * D;
    const float* arow = agg + (nbase + mrow) * D;
    v2f afrag[16];
#pragma unroll
    for (int kc = 0; kc < 16; ++kc) {
      int k0 = kc * 4 + khalf;
      afrag[kc].x = epsv * hrow[k0]     + arow[k0];
      afrag[kc].y = epsv * hrow[k0 + 1] + arow[k0 + 1];
    }
#pragma unroll
    for (int j = 0; j < 4; ++j) {
      int cg = j * 16 + mrow;       // output column of this lane
      v8f c = {0.f, 0.f, 0.f, 0.f, 0.f, 0.f, 0.f, 0.f};
#pragma unroll
      for (int kc = 0; kc < 16; ++kc) {
        int k0 = kc * 4 + khalf;
        v2f b;
        b.x = sW[k0 * D + cg];
        b.y = sW[(k0 + 1) * D + cg];
        c = __builtin_amdgcn_wmma_f32_16x16x4_f32(false, afrag[kc], false, b,
                                                  (short)0, c, false, false);
      }
      // C layout: VGPR v holds row v (lanes 0-15) / row v+8 (lanes 16-31)
      float* yb = y + (nbase + ((lane >> 4) << 3)) * D + cg;
      float s = 0.0f, q = 0.0f;
#pragma unroll
      for (int v = 0; v < 8; ++v) {
        float val = c[v];
        yb[v * D] = val;
        s += val;
        q += val * val;
      }
      // lanes L and L+16 hold the same column -> pair-combine then LDS-reduce
      s += __shfl_down(s, 16, 32);
      q += __shfl_down(q, 16, 32);
      if (lane < 16) {
        atomicAdd(&sSum[cg], s);
        atomicAdd(&sSq[cg], q);
      }
    }
  }
  __syncthreads();
  if (tid < D) {
    unsafeAtomicAdd(&colSum[tid], sSum[tid]);
    unsafeAtomicAdd(&colSq[tid], sSq[tid]);
  }
}

// ---------------------------------------------------------------------------
// pass2: t = relu(y*scale1 + shift1) ;  z2 = t @ W2  (bias absorbed by BN2)
// Same WMMA structure; BN1+ReLU fused into the A-fragment load.
// ---------------------------------------------------------------------------
__global__ __launch_bounds__(256) void k_pass2(
    const float* __restrict__ y, const float* __restrict__ W,
    const float* __restrict__ scale, const float* __restrict__ shiftv,
    float* __restrict__ z2, float* __restrict__ colSum, float* __restrict__ colSq,
    int nNodes) {
  __shared__ float sW[D * D];
  __shared__ float sSc[D];
  __shared__ float sSh[D];
  __shared__ float sSum[D];
  __shared__ float sSq[D];
  int tid = threadIdx.x;
  for (int i = tid; i < D * D; i += 256) sW[i] = W[i];
  if (tid < D) {
    sSc[tid] = scale[tid];
    sSh[tid] = shiftv[tid];
    sSum[tid] = 0.0f;
    sSq[tid] = 0.0f;
  }
  __syncthreads();

  int lane  = tid & 31;
  int wave  = tid >> 5;
  int mrow  = lane & 15;
  int khalf = (lane >> 4) << 1;
  int nbase = blockIdx.x * 128 + wave * 16;

  if (nbase < nNodes) {
    const float* yrow = y + (nbase + mrow) * D;
    v2f afrag[16];
#pragma unroll
    for (int kc = 0; kc < 16; ++kc) {
      int k0 = kc * 4 + khalf;
      afrag[kc].x = fmaxf(fmaf(yrow[k0],     sSc[k0],     sSh[k0]),     0.0f);
      afrag[kc].y = fmaxf(fmaf(yrow[k0 + 1], sSc[k0 + 1], sSh[k0 + 1]), 0.0f);
    }
#pragma unroll
    for (int j = 0; j < 4; ++j) {
      int cg = j * 16 + mrow;
      v8f c = {0.f, 0.f, 0.f, 0.f, 0.f, 0.f, 0.f, 0.f};
#pragma unroll
      for (int kc = 0; kc < 16; ++kc) {
        int k0 = kc * 4 + khalf;
        v2f b;
        b.x = sW[k0 * D + cg];
        b.y = sW[(k0 + 1) * D + cg];
        c = __builtin_amdgcn_wmma_f32_16x16x4_f32(false, afrag[kc], false, b,
                                                  (short)0, c, false, false);
      }
      float* zb = z2 + (nbase + ((lane >> 4) << 3)) * D + cg;
      float s = 0.0f, q = 0.0f;
#pragma unroll
      for (int v = 0; v < 8; ++v) {
        float val = c[v];
        zb[v * D] = val;
        s += val;
        q += val * val;
      }
      s += __shfl_down(s, 16, 32);
      q += __shfl_down(q, 16, 32);
      if (lane < 16) {
        atomicAdd(&sSum[cg], s);
        atomicAdd(&sSq[cg], q);
      }
    }
  }
  __syncthreads();
  if (tid < D) {
    unsafeAtomicAdd(&colSum[tid], sSum[tid]);
    unsafeAtomicAdd(&colSq[tid], sSq[tid]);
  }
}

// ---------------------------------------------------------------------------
// BN finalize: scale = gamma*rsqrt(var+eps); shift = beta - mu*scale
// ---------------------------------------------------------------------------
__global__ void k_bnfin(const float* __restrict__ sum, const float* __restrict__ sq,
                        const float* __restrict__ gamma, const float* __restrict__ beta,
                        float* __restrict__ scale, float* __restrict__ shift, float invN) {
  int c = threadIdx.x;
  float mu  = sum[c] * invN;
  float var = sq[c] * invN - mu * mu;
  float rs  = rsqrtf(var + 1e-5f);
  float scv = gamma[c] * rs;
  scale[c] = scv;
  shift[c] = fmaf(-mu, scv, beta[c]);
}

// h = relu(z*scale2 + shift2)
__global__ __launch_bounds__(256) void k_bnrelu(const float* __restrict__ z,
                                                const float* __restrict__ sc,
                                                const float* __restrict__ sh,
                                                float* __restrict__ hout, int total) {
  int i = blockIdx.x * 256 + threadIdx.x;
  if (i < total) {
    int c = i & 63;
    hout[i] = fmaxf(fmaf(z[i], sc[c], sh[c]), 0.0f);
  }
}

// pooled[batch[n]] += h[n]
__global__ __launch_bounds__(256) void k_pool(const float* __restrict__ h,
                                              const int* __restrict__ batch,
                                              float* __restrict__ pooled, int total) {
  int i = blockIdx.x * 256 + threadIdx.x;
  if (i < total) {
    int n = i >> 6, d = i & 63;
    unsafeAtomicAdd(&pooled[batch[n] * D + d], h[i]);
  }
}

// out[g] = relu(pooled[g] @ Wf1 + bf1) @ Wf2 + bf2    (2D = 128 = blockDim)
__global__ __launch_bounds__(128) void k_head(const float* __restrict__ pooled,
                                              const float* __restrict__ Wf1,
                                              const float* __restrict__ bf1,
                                              const float* __restrict__ Wf2,
                                              const float* __restrict__ bf2,
                                              float* __restrict__ out, int NT) {
  __shared__ float sp[D];
  __shared__ float red[128];
  int g = blockIdx.x, j = threadIdx.x;
  if (j < D) sp[j] = pooled[g * D + j];
  __syncthreads();
  float acc = bf1[j];
  for (int d = 0; d < D; ++d) acc = fmaf(sp[d], Wf1[d * 2 * D + j], acc);
  acc = fmaxf(acc, 0.0f);
  for (int t = 0; t < NT; ++t) {
    red[j] = acc * Wf2[j * NT + t];
    __syncthreads();
    for (int s = 64; s > 0; s >>= 1) {
      if (j < s) red[j] += red[j + s];
      __syncthreads();
    }
    if (j == 0) out[g * NT + t] = red[0] + bf2[t];
    __syncthreads();
  }
}

// ---------------------------------------------------------------------------
extern "C" void kernel_launch(void* const* d_in, const int* in_sizes, int n_in,
                              void* d_out, int out_size, void* d_ws, size_t ws_size,
                              hipStream_t stream) {
  const int*   x     = (const int*)d_in[0];
  const int*   eidx  = (const int*)d_in[1];   // [2, E]: src row then dst row
  const int*   eattr = (const int*)d_in[2];
  const int*   batch = (const int*)d_in[3];
  const float* femb  = (const float*)d_in[4];
  const float* eemb  = (const float*)d_in[5];
  const float* eps   = (const float*)d_in[6];
  const float* W1    = (const float*)d_in[7];
  // b1 (d_in[8]) and b2 (d_in[12]) are exactly cancelled by batch-stat BN.
  const float* g1    = (const float*)d_in[9];
  const float* be1   = (const float*)d_in[10];
  const float* W2    = (const float*)d_in[11];
  const float* g2    = (const float*)d_in[13];
  const float* be2   = (const float*)d_in[14];
  const float* Wf1   = (const float*)d_in[15];
  const float* bf1   = (const float*)d_in[16];
  const float* Wf2   = (const float*)d_in[17];
  const float* bf2   = (const float*)d_in[18];

  int N  = in_sizes[0];
  int E  = in_sizes[2];
  int L  = in_sizes[6];
  int NT = in_sizes[18];
  int G  = out_size / NT;
  int NE = in_sizes[5] / (L * D);

  // workspace layout (floats)
  float* hbuf   = (float*)d_ws;                    // N*D   (node features)
  float* aggbuf = hbuf + (size_t)N * D;            // N*D   (agg, reused as z2)
  float* ybuf   = aggbuf + (size_t)N * D;          // N*D   (pre-BN1 activations)
  float* pooled = ybuf + (size_t)N * D;            // G*D
  float* stSum  = pooled + (size_t)G * D;          // D
  float* stSq   = stSum + D;                       // D
  float* stSc   = stSq + D;                        // D
  float* stSh   = stSc + D;                        // D

  const int totND = N * D;
  const int totED = E * D;
  dim3 blk(256);
  int gridND   = (totND + 255) / 256;
  int gridED   = (totED + 255) / 256;
  int gridGEMM = (N + 127) / 128;                  // 128 nodes per block (8 waves x 16)
  float invN = 1.0f / (float)N;

  k_embed<<<gridND, blk, 0, stream>>>(x, femb, hbuf, totND);

  for (int l = 0; l < L; ++l) {
    hipMemsetAsync(aggbuf, 0, (size_t)totND * sizeof(float), stream);
    k_message<<<gridED, blk, 0, stream>>>(hbuf, eidx, eidx + E, eattr,
                                          eemb + (size_t)l * NE * D, aggbuf, totED, NE);

    hipMemsetAsync(stSum, 0, (size_t)2 * D * sizeof(float), stream);
    k_pass1<<<gridGEMM, blk, 0, stream>>>(hbuf, aggbuf, W1 + (size_t)l * D * D, eps, l,
                                          ybuf, stSum, stSq, N);
    k_bnfin<<<1, D, 0, stream>>>(stSum, stSq, g1 + (size_t)l * D, be1 + (size_t)l * D,
                                 stSc, stSh, invN);

    hipMemsetAsync(stSum, 0, (size_t)2 * D * sizeof(float), stream);
    k_pass2<<<gridGEMM, blk, 0, stream>>>(ybuf, W2 + (size_t)l * D * D, stSc, stSh,
                                          aggbuf, stSum, stSq, N);
    k_bnfin<<<1, D, 0, stream>>>(stSum, stSq, g2 + (size_t)l * D, be2 + (size_t)l * D,
                                 stSc, stSh, invN);

    k_bnrelu<<<gridND, blk, 0, stream>>>(aggbuf, stSc, stSh, hbuf, totND);
  }

  hipMemsetAsync(pooled, 0, (size_t)G * D * sizeof(float), stream);
  k_pool<<<gridND, blk, 0, stream>>>(hbuf, batch, pooled, totND);
  k_head<<<G, 128, 0, stream>>>(pooled, Wf1, bf1, Wf2, bf2, (float*)d_out, NT);
}